// HardL1ACELoss_47777216200803
// MI455X (gfx1250) — compile-verified
//
#include <hip/hip_runtime.h>
#include <hip/hip_bf16.h>
#include <stdint.h>

#define NBINS   20
#define BLOCK   256
#define NWAVES  (BLOCK / 32)        // wave32 on gfx1250
#define HSLOTS  (3 * NBINS)         // count / sum_p / sum_t per wave
#define EPS     1e-8

// ---------------------------------------------------------------------------
// CDNA5 async global->LDS copy (ASYNCcnt path, gfx1250-specific).
// Each lane copies 16 bytes from global address sbase plus voff to the LDS address in lds_byte_addr.
// ---------------------------------------------------------------------------
__device__ __forceinline__ void async_cp16(unsigned lds_byte_addr,
                                           const void* sbase,
                                           unsigned voff) {
  asm volatile("global_load_async_to_lds_b128 %0, %1, %2"
               :
               : "v"(lds_byte_addr), "v"(voff), "s"(sbase)
               : "memory");
}
__device__ __forceinline__ void wait_async_le2() {
  asm volatile("s_wait_asynccnt 0x2" ::: "memory");
}
__device__ __forceinline__ void wait_async_0() {
  asm volatile("s_wait_asynccnt 0x0" ::: "memory");
}

// Per-element histogram update into this wave's private LDS histogram.
// bin = floor(p * 20); p in [0,1) => truncation == floor. Out-of-range -> drop
// (matches reference overflow-segment behavior).
__device__ __forceinline__ void accum(float* lh, float p, float t) {
  int b = (int)(p * (float)NBINS);
  if ((unsigned)b < (unsigned)NBINS) {
    __hip_atomic_fetch_add(lh + b,             1.0f, __ATOMIC_RELAXED, __HIP_MEMORY_SCOPE_WORKGROUP);
    __hip_atomic_fetch_add(lh + NBINS + b,     p,    __ATOMIC_RELAXED, __HIP_MEMORY_SCOPE_WORKGROUP);
    __hip_atomic_fetch_add(lh + 2 * NBINS + b, t,    __ATOMIC_RELAXED, __HIP_MEMORY_SCOPE_WORKGROUP);
  }
}

// ---------------------------------------------------------------------------
// Kernel 0: zero the 64-double workspace accumulators (ws is poisoned 0xAA).
// ---------------------------------------------------------------------------
__global__ void ace_zero_kernel(double* __restrict__ ws) {
  int i = threadIdx.x;
  if (i < 64) ws[i] = 0.0;
}

// ---------------------------------------------------------------------------
// Kernel 1: streaming histogram.
//   - double-buffered async global->LDS staging (16 KB LDS) when addresses
//     fit 32-bit voffsets (they do for this problem: 134 MB per tensor)
//   - per-wave LDS histograms (ds_add_f32), block reduce, f64 global atomics
// ---------------------------------------------------------------------------
__global__ void __launch_bounds__(BLOCK)
ace_hist_kernel(const float* __restrict__ preds,
                const float* __restrict__ targs,
                double* __restrict__ ws,
                long long n,
                int use_async) {
  __shared__ float4 stg_p[2][BLOCK];
  __shared__ float4 stg_t[2][BLOCK];
  __shared__ float  lhist[NWAVES * HSLOTS];

  const int tid = threadIdx.x;
  float* lh = &lhist[(tid >> 5) * HSLOTS];

  for (int i = tid; i < NWAVES * HSLOTS; i += BLOCK) lhist[i] = 0.0f;
  __syncthreads();

  const long long nq     = n >> 2;                        // float4 quads
  const long long stride = (long long)gridDim.x * BLOCK;
  const long long g0     = (long long)blockIdx.x * BLOCK + tid;

  if (use_async) {
    const long long it = nq / stride;   // grid-uniform trip count -> no EXEC
                                        // divergence around wave-level waits
    // LDS byte addresses of this thread's staging slots (low 32 bits of the
    // flat address of an LDS object == DS-relative byte offset).
    unsigned lp[2], lt[2];
    lp[0] = (unsigned)(uintptr_t)&stg_p[0][tid];
    lp[1] = (unsigned)(uintptr_t)&stg_p[1][tid];
    lt[0] = (unsigned)(uintptr_t)&stg_t[0][tid];
    lt[1] = (unsigned)(uintptr_t)&stg_t[1][tid];

    if (it > 0) {
      unsigned voff0 = (unsigned)(g0 * 16ll);
      async_cp16(lp[0], preds, voff0);
      async_cp16(lt[0], targs, voff0);
      int buf = 0;
      for (long long k = 0; k < it; ++k) {
        if (k + 1 < it) {
          unsigned voff = (unsigned)((g0 + (k + 1) * stride) * 16ll);
          async_cp16(lp[buf ^ 1], preds, voff);   // prefetch next tile
          async_cp16(lt[buf ^ 1], targs, voff);
          wait_async_le2();                       // current tile (oldest 2) done
        } else {
          wait_async_0();
        }
        float4 p4 = stg_p[buf][tid];
        float4 t4 = stg_t[buf][tid];
        accum(lh, p4.x, t4.x);
        accum(lh, p4.y, t4.y);
        accum(lh, p4.z, t4.z);
        accum(lh, p4.w, t4.w);
        buf ^= 1;
      }
    }
    // remainder quads (rem < stride): one direct b128 load per thread
    const long long done = it * stride;
    if (g0 < (nq - done)) {
      float4 p4 = ((const float4*)preds)[done + g0];
      float4 t4 = ((const float4*)targs)[done + g0];
      accum(lh, p4.x, t4.x);
      accum(lh, p4.y, t4.y);
      accum(lh, p4.z, t4.z);
      accum(lh, p4.w, t4.w);
    }
  } else {
    // fallback for tensors too large for 32-bit voffsets: plain grid-stride
    for (long long q = g0; q < nq; q += stride) {
      float4 p4 = ((const float4*)preds)[q];
      float4 t4 = ((const float4*)targs)[q];
      accum(lh, p4.x, t4.x);
      accum(lh, p4.y, t4.y);
      accum(lh, p4.z, t4.z);
      accum(lh, p4.w, t4.w);
    }
  }

  // scalar tail (n % 4 elements)
  if (g0 == 0) {
    for (long long i = (nq << 2); i < n; ++i) accum(lh, preds[i], targs[i]);
  }

  __syncthreads();
  // reduce 8 per-wave histograms, accumulate globally in f64 (exact counts,
  // tight Σp-Σt cancellation across 2048 blocks)
  for (int i = tid; i < HSLOTS; i += BLOCK) {
    float s = 0.0f;
    for (int w = 0; w < NWAVES; ++w) s += lhist[w * HSLOTS + i];
    __hip_atomic_fetch_add(&ws[i], (double)s, __ATOMIC_RELAXED, __HIP_MEMORY_SCOPE_AGENT);
  }
}

// ---------------------------------------------------------------------------
// Kernel 2: scalar epilogue, ACE = sum(|Σp-Σt|/count over valid bins)/(V+eps)
// ---------------------------------------------------------------------------
__global__ void ace_final_kernel(const double* __restrict__ ws,
                                 float* __restrict__ out) {
  if (threadIdx.x == 0) {
    double ace = 0.0, valid = 0.0;
    for (int b = 0; b < NBINS; ++b) {
      double c  = ws[b];
      double sp = ws[NBINS + b];
      double st = ws[2 * NBINS + b];
      if (c > 0.0) {
        ace += fabs(sp - st) / c;
        valid += 1.0;
      }
    }
    out[0] = (float)(ace / (valid + EPS));
  }
}

// ---------------------------------------------------------------------------
extern "C" void kernel_launch(void* const* d_in, const int* in_sizes, int n_in,
                              void* d_out, int out_size, void* d_ws, size_t ws_size,
                              hipStream_t stream) {
  (void)n_in; (void)out_size; (void)ws_size;
  const float* preds = (const float*)d_in[0];
  const float* targs = (const float*)d_in[1];
  double* ws  = (double*)d_ws;       // needs 64 doubles = 512 B of scratch
  float*  out = (float*)d_out;

  const long long n  = (long long)in_sizes[0];
  const long long nq = n >> 2;

  int blocks = (int)((nq + BLOCK - 1) / BLOCK);
  if (blocks > 2048) blocks = 2048;
  if (blocks < 1) blocks = 1;

  // 32-bit byte voffset must cover the whole tensor for the async path
  const int use_async = ((unsigned long long)n * 4ull) < (1ull << 31) ? 1 : 0;

  ace_zero_kernel<<<1, 64, 0, stream>>>(ws);
  ace_hist_kernel<<<blocks, BLOCK, 0, stream>>>(preds, targs, ws, n, use_async);
  ace_final_kernel<<<1, 32, 0, stream>>>(ws, out);
}